// GATTemporalEncoder_72241349919130
// MI455X (gfx1250) — compile-verified
//
#include <hip/hip_runtime.h>
#include <math.h>

#define NN 10000
#define TT 6
#define FF 64
#define HD 64
#define NH 4

typedef __attribute__((ext_vector_type(2))) float v2f;
typedef __attribute__((ext_vector_type(8))) float v8f;

// ---------------------------------------------------------------------------
// float atomic max via signed/unsigned int monotonicity trick
__device__ __forceinline__ void atomic_max_f32(float* addr, float val) {
  if (val >= 0.0f) atomicMax((int*)addr, __float_as_int(val));
  else             atomicMin((unsigned int*)addr, (unsigned int)__float_as_int(val));
}

__device__ __forceinline__ float elu1(float v) { return v > 0.0f ? v : __expf(v) - 1.0f; }
__device__ __forceinline__ float sigm(float v) { return 1.0f / (1.0f + __expf(-v)); }

// ---------------------------------------------------------------------------
// C[nrows x mcols] = A[nrows x K] @ B   (f32, V_WMMA_F32_16X16X4_F32)
// A row stride lda. If !BT: B is [K x mcols], row stride ldb.
// If BT: B is [mcols x K] row-major (logical B[k][m] = B[m*ldb + k]).
// One wave per 16x16 C tile; K compile-time -> fully unrolled, branch-free
// inner loop (all loads issued up front, 16 back-to-back WMMAs).
template <bool BT, int K>
__global__ void wmma_gemm_f32(const float* __restrict__ A, int lda,
                              const float* __restrict__ B, int ldb,
                              float* __restrict__ C, int ldc,
                              int nrows, int mcols) {
  const int wave = (blockIdx.x * blockDim.x + threadIdx.x) >> 5;
  const int tiles_m = mcols >> 4;
  const int tiles = (nrows >> 4) * tiles_m;
  if (wave >= tiles) return;                 // wave-uniform: EXEC all-ones below
  const int row0 = (wave / tiles_m) << 4;
  const int col0 = (wave % tiles_m) << 4;
  const int lane = threadIdx.x & 31;
  const int mn = lane & 15;                  // M index (A) / N index (B,C)
  const int kb = (lane >> 4) << 1;           // lanes 16-31 carry K+2,K+3

  const float* arow = A + (size_t)(row0 + mn) * lda;
  v2f av[K / 4], bv[K / 4];
#pragma unroll
  for (int i = 0; i < K / 4; ++i) {
    const int k = 4 * i;
    av[i].x = arow[k + kb];
    av[i].y = arow[k + kb + 1];
    if (BT) {
      const float* bp = B + (size_t)(col0 + mn) * ldb + k + kb;
      bv[i].x = bp[0];
      bv[i].y = bp[1];
    } else {
      bv[i].x = B[(size_t)(k + kb) * ldb + col0 + mn];
      bv[i].y = B[(size_t)(k + kb + 1) * ldb + col0 + mn];
    }
  }

  v8f c = {};
#pragma unroll
  for (int i = 0; i < K / 4; ++i) {
    // 8 args: (neg_a, A, neg_b, B, c_mod, C, reuse_a, reuse_b)
    c = __builtin_amdgcn_wmma_f32_16x16x4_f32(false, av[i], false, bv[i],
                                              (short)0, c, false, false);
  }

  // C/D layout: VGPR r -> M = r + 8*(lane>=16), N = lane&15
  float* crow = C + (size_t)(row0 + ((lane >> 4) << 3)) * ldc + col0 + mn;
#pragma unroll
  for (int r = 0; r < 8; ++r) crow[(size_t)r * ldc] = c[r];
}

// ---------------------------------------------------------------------------
// per-node attention scores: s_src[n,h] = sum_c xw[n,h,c]*a_src[h,c] (same dst)
__global__ void attn_scores(const float* __restrict__ xw,
                            const float* __restrict__ a_src,
                            const float* __restrict__ a_dst,
                            float* __restrict__ ssrc, float* __restrict__ sdst,
                            int n, int C) {
  int i = blockIdx.x * blockDim.x + threadIdx.x;
  if (i >= n * NH) return;
  int node = i >> 2, h = i & 3;
  const float* xp = xw + (size_t)node * NH * C + h * C;
  const float* as = a_src + h * C;
  const float* ad = a_dst + h * C;
  float s1 = 0.0f, s2 = 0.0f;
  for (int c = 0; c < C; ++c) { float v = xp[c]; s1 += v * as[c]; s2 += v * ad[c]; }
  ssrc[i] = s1;
  sdst[i] = s2;
}

// init segment-softmax state + zero aggregation buffer
__global__ void seg_init(float* __restrict__ smax, float* __restrict__ ssum,
                         float* __restrict__ agg, int n) {
  int i = blockIdx.x * blockDim.x + threadIdx.x;
  if (i >= n * HD) return;
  agg[i] = 0.0f;
  if (i < n * NH) { smax[i] = -__builtin_inff(); ssum[i] = 0.0f; }
}

// pass 1: logits + leaky relu + segment max over dst
__global__ void edge_logits(const int* __restrict__ e_src, const int* __restrict__ e_dst,
                            int E, int Ep,
                            const float* __restrict__ ssrc, const float* __restrict__ sdst,
                            float* __restrict__ ex, float* __restrict__ smax) {
  int i = blockIdx.x * blockDim.x + threadIdx.x;
  if (i >= Ep * NH) return;
  int e = i >> 2, h = i & 3;
  int s = e < E ? e_src[e] : e - E;
  int d = e < E ? e_dst[e] : e - E;
  float al = ssrc[s * NH + h] + sdst[d * NH + h];
  al = al > 0.0f ? al : 0.2f * al;
  ex[i] = al;
  atomic_max_f32(&smax[d * NH + h], al);
}

// pass 2: exp(al - max) + segment sum
__global__ void edge_expsum(const int* __restrict__ e_dst, int E, int Ep,
                            float* __restrict__ ex, const float* __restrict__ smax,
                            float* __restrict__ ssum) {
  int i = blockIdx.x * blockDim.x + threadIdx.x;
  if (i >= Ep * NH) return;
  int e = i >> 2, h = i & 3;
  int d = e < E ? e_dst[e] : e - E;
  float v = __expf(ex[i] - smax[d * NH + h]);
  ex[i] = v;
  atomicAdd(&ssum[d * NH + h], v);
}

// pass 3a (layer 0, concat, C=16): agg[dst, h*16+c] += xw[src, h*16+c] * alpha
__global__ void edge_agg_concat(const int* __restrict__ e_src, const int* __restrict__ e_dst,
                                int E, int Ep,
                                const float* __restrict__ ex, const float* __restrict__ ssum,
                                const float* __restrict__ xw, float* __restrict__ agg) {
  long long i = (long long)blockIdx.x * blockDim.x + threadIdx.x;
  if (i >= (long long)Ep * HD) return;
  int e = (int)(i >> 6), j = (int)(i & 63), h = j >> 4;
  int s = e < E ? e_src[e] : e - E;
  int d = e < E ? e_dst[e] : e - E;
  float alpha = ex[e * NH + h] / (ssum[d * NH + h] + 1e-16f);
  atomicAdd(&agg[(size_t)d * HD + j], xw[(size_t)s * HD + j] * alpha);
}

// pass 3b (layer 1, head-mean folded in, C=64): agg[dst,c] += 0.25*sum_h xw[src,h,c]*alpha_h
__global__ void edge_agg_mean(const int* __restrict__ e_src, const int* __restrict__ e_dst,
                              int E, int Ep,
                              const float* __restrict__ ex, const float* __restrict__ ssum,
                              const float* __restrict__ xw, float* __restrict__ agg,
                              float* __restrict__ alpha_out) {
  long long i = (long long)blockIdx.x * blockDim.x + threadIdx.x;
  if (i >= (long long)Ep * HD) return;
  int e = (int)(i >> 6), c = (int)(i & 63);
  int s = e < E ? e_src[e] : e - E;
  int d = e < E ? e_dst[e] : e - E;
  float acc = 0.0f;
#pragma unroll
  for (int h = 0; h < NH; ++h) {
    float alpha = ex[e * NH + h] / (ssum[d * NH + h] + 1e-16f);
    acc += xw[(size_t)s * 256 + h * 64 + c] * alpha;
    if (alpha_out && c == h) alpha_out[(size_t)e * NH + h] = alpha;
  }
  atomicAdd(&agg[(size_t)d * HD + c], 0.25f * acc);
}

// h = elu(agg + bias), in place
__global__ void bias_elu(float* __restrict__ x, const float* __restrict__ b, int total) {
  int i = blockIdx.x * blockDim.x + threadIdx.x;
  if (i >= total) return;
  x[i] = elu1(x[i] + b[i & 63]);
}

// out = layernorm(elu(agg + bias)) * g + beta ; one wave per node, 2 vals/lane
__global__ void bias_elu_ln(const float* __restrict__ agg, const float* __restrict__ b,
                            const float* __restrict__ g, const float* __restrict__ beta,
                            float* __restrict__ out, int n) {
  int wave = (blockIdx.x * blockDim.x + threadIdx.x) >> 5;
  if (wave >= n) return;
  int lane = threadIdx.x & 31;
  float v0 = elu1(agg[(size_t)wave * HD + lane] + b[lane]);
  float v1 = elu1(agg[(size_t)wave * HD + 32 + lane] + b[32 + lane]);
  float s = v0 + v1;
#pragma unroll
  for (int o = 16; o > 0; o >>= 1) s += __shfl_xor(s, o, 32);
  float mu = s * (1.0f / 64.0f);
  float d0 = v0 - mu, d1 = v1 - mu;
  float q = d0 * d0 + d1 * d1;
#pragma unroll
  for (int o = 16; o > 0; o >>= 1) q += __shfl_xor(q, o, 32);
  float rstd = rsqrtf(q * (1.0f / 64.0f) + 1e-5f);
  out[(size_t)wave * HD + lane]      = d0 * rstd * g[lane] + beta[lane];
  out[(size_t)wave * HD + 32 + lane] = d1 * rstd * g[32 + lane] + beta[32 + lane];
}

// GRU gate fusion: h = (1-z)*g + z*h
__global__ void gru_gates(const float* __restrict__ gi, const float* __restrict__ gh,
                          const float* __restrict__ b_ih, const float* __restrict__ b_hh,
                          float* __restrict__ h, int n) {
  int i = blockIdx.x * blockDim.x + threadIdx.x;
  if (i >= n * HD) return;
  int node = i >> 6, j = i & 63;
  const float* gip = gi + (size_t)node * 192;
  const float* ghp = gh + (size_t)node * 192;
  float r = sigm(gip[j] + b_ih[j] + ghp[j] + b_hh[j]);
  float z = sigm(gip[64 + j] + b_ih[64 + j] + ghp[64 + j] + b_hh[64 + j]);
  float g = tanhf(gip[128 + j] + b_ih[128 + j] + r * (ghp[128 + j] + b_hh[128 + j]));
  h[i] = (1.0f - z) * g + z * h[i];
}

__global__ void zero_f32(float* __restrict__ p, int n) {
  int i = blockIdx.x * blockDim.x + threadIdx.x;
  if (i < n) p[i] = 0.0f;
}

// attn_edge_index (with self loops) written as floats after alpha block
__global__ void write_edge_index(const int* __restrict__ e_src, const int* __restrict__ e_dst,
                                 int E, int Ep, float* __restrict__ out) {
  int i = blockIdx.x * blockDim.x + threadIdx.x;
  if (i >= Ep) return;
  out[i]      = (float)(i < E ? e_src[i] : i - E);
  out[Ep + i] = (float)(i < E ? e_dst[i] : i - E);
}

// ---------------------------------------------------------------------------
static inline int blks(long long n) { return (int)((n + 255) / 256); }

extern "C" void kernel_launch(void* const* d_in, const int* in_sizes, int n_in,
                              void* d_out, int out_size, void* d_ws, size_t ws_size,
                              hipStream_t stream) {
  const float* x      = (const float*)d_in[0];
  const int*   eidx   = (const int*)d_in[1];
  const float* W0     = (const float*)d_in[2];
  const float* a_src0 = (const float*)d_in[3];
  const float* a_dst0 = (const float*)d_in[4];
  const float* b0     = (const float*)d_in[5];
  const float* W1     = (const float*)d_in[6];
  const float* a_src1 = (const float*)d_in[7];
  const float* a_dst1 = (const float*)d_in[8];
  const float* b1     = (const float*)d_in[9];
  const float* ln_g   = (const float*)d_in[10];
  const float* ln_b   = (const float*)d_in[11];
  const float* W_ih   = (const float*)d_in[12];
  const float* W_hh   = (const float*)d_in[13];
  const float* b_ih   = (const float*)d_in[14];
  const float* b_hh   = (const float*)d_in[15];

  const int E  = in_sizes[1] / 2;
  const int Ep = E + NN;
  const int* e_src = eidx;
  const int* e_dst = eidx + E;

  float* ws = (float*)d_ws;
  size_t off = 0;
  float* XW1  = ws + off; off += (size_t)NN * 256;
  float* XW0  = ws + off; off += (size_t)NN * HD;
  float* SSRC = ws + off; off += (size_t)NN * NH;
  float* SDST = ws + off; off += (size_t)NN * NH;
  float* SMAX = ws + off; off += (size_t)NN * NH;
  float* SSUM = ws + off; off += (size_t)NN * NH;
  float* EX   = ws + off; off += (size_t)Ep * NH;
  float* AGG  = ws + off; off += (size_t)NN * HD;   // layer0 agg -> h0 (in place elu)
  float* AGG1 = ws + off; off += (size_t)NN * HD;
  float* HS   = ws + off; off += (size_t)TT * NN * HD;
  float* GI   = ws + off; off += (size_t)NN * 192;
  float* GH   = ws + off; off += (size_t)NN * 192;
  float* H    = ws + off; off += (size_t)NN * HD;
  (void)ws_size; (void)n_in; (void)out_size;

  float* out_hT    = (float*)d_out;
  float* out_alpha = out_hT + (size_t)NN * HD;
  float* out_eidx  = out_alpha + (size_t)Ep * NH;

  for (int t = 0; t < TT; ++t) {
    // ---- GAT layer 0: xw0 = x[:,t,:] @ W0  (A row stride T*F) ----
    {
      int tiles = (NN / 16) * (HD / 16);
      wmma_gemm_f32<false, 64><<<blks((long long)tiles * 32), 256, 0, stream>>>(
          x + (size_t)t * FF, TT * FF, W0, HD, XW0, HD, NN, HD);
    }
    attn_scores<<<blks(NN * NH), 256, 0, stream>>>(XW0, a_src0, a_dst0, SSRC, SDST, NN, HD / NH);
    seg_init<<<blks(NN * HD), 256, 0, stream>>>(SMAX, SSUM, AGG, NN);
    edge_logits<<<blks((long long)Ep * NH), 256, 0, stream>>>(e_src, e_dst, E, Ep, SSRC, SDST, EX, SMAX);
    edge_expsum<<<blks((long long)Ep * NH), 256, 0, stream>>>(e_dst, E, Ep, EX, SMAX, SSUM);
    edge_agg_concat<<<blks((long long)Ep * HD), 256, 0, stream>>>(e_src, e_dst, E, Ep, EX, SSUM, XW0, AGG);
    bias_elu<<<blks(NN * HD), 256, 0, stream>>>(AGG, b0, NN * HD);

    // ---- GAT layer 1: xw1 = h0 @ W1 ----
    {
      int tiles = (NN / 16) * (256 / 16);
      wmma_gemm_f32<false, 64><<<blks((long long)tiles * 32), 256, 0, stream>>>(
          AGG, HD, W1, 256, XW1, 256, NN, 256);
    }
    attn_scores<<<blks(NN * NH), 256, 0, stream>>>(XW1, a_src1, a_dst1, SSRC, SDST, NN, HD);
    seg_init<<<blks(NN * HD), 256, 0, stream>>>(SMAX, SSUM, AGG1, NN);
    edge_logits<<<blks((long long)Ep * NH), 256, 0, stream>>>(e_src, e_dst, E, Ep, SSRC, SDST, EX, SMAX);
    edge_expsum<<<blks((long long)Ep * NH), 256, 0, stream>>>(e_dst, E, Ep, EX, SMAX, SSUM);
    edge_agg_mean<<<blks((long long)Ep * HD), 256, 0, stream>>>(
        e_src, e_dst, E, Ep, EX, SSUM, XW1, AGG1, (t == TT - 1) ? out_alpha : (float*)nullptr);
    bias_elu_ln<<<blks((long long)NN * 32), 256, 0, stream>>>(
        AGG1, b1, ln_g, ln_b, HS + (size_t)t * NN * HD, NN);
  }

  // ---- GRU over T steps ----
  zero_f32<<<blks(NN * HD), 256, 0, stream>>>(H, NN * HD);
  {
    int tiles = (NN / 16) * (192 / 16);
    for (int t = 0; t < TT; ++t) {
      wmma_gemm_f32<true, 64><<<blks((long long)tiles * 32), 256, 0, stream>>>(
          HS + (size_t)t * NN * HD, HD, W_ih, HD, GI, 192, NN, 192);
      wmma_gemm_f32<true, 64><<<blks((long long)tiles * 32), 256, 0, stream>>>(
          H, HD, W_hh, HD, GH, 192, NN, 192);
      gru_gates<<<blks(NN * HD), 256, 0, stream>>>(GI, GH, b_ih, b_hh, H, NN);
    }
  }

  hipMemcpyAsync(out_hT, H, (size_t)NN * HD * sizeof(float), hipMemcpyDeviceToDevice, stream);
  write_edge_index<<<blks(Ep), 256, 0, stream>>>(e_src, e_dst, E, Ep, out_eidx);
}